// DWT1D_49349174231513
// MI455X (gfx1250) — compile-verified
//
#include <hip/hip_runtime.h>
#include <hip/hip_bf16.h>

typedef __attribute__((ext_vector_type(2))) float v2f;
typedef __attribute__((ext_vector_type(8))) float v8f;

#define C_CH      64     // channels (contiguous, innermost)
#define TILE_I    32     // output rows (per filter) per block
#define ROWS      72     // staged input rows (need 70 + 2 K-pad)
#define RSTRIDE   72     // LDS row stride in floats (64 + 8 pad -> half-waves hit disjoint banks)
#define NCHUNK    10     // K chunks of 4 covering banded 16x40 filter matrix

__device__ const float kDecLo[8] = {
    -0.010597401784997278f,  0.032883011666982945f,  0.030841381835986965f, -0.18703481171888114f,
    -0.02798376941698385f,   0.6308807679295904f,    0.7148465705525415f,    0.23037781330885523f };
__device__ const float kDecHi[8] = {
    -0.23037781330885523f,   0.7148465705525415f,   -0.6308807679295904f,   -0.02798376941698385f,
     0.18703481171888114f,   0.030841381835986965f,  0.032883011666982945f, -0.010597401784997278f };

// One DWT level: in (32, N, 64) f32 -> outA (32, N/2, 64), outD (32, N/2, 64).
// out[b,i,c] = sum_j filt[j] * in[b, (2i+j) mod N, c]
__global__ __launch_bounds__(256) void dwt_level_wmma(const float* __restrict__ in,
                                                      float* __restrict__ outA,
                                                      float* __restrict__ outD,
                                                      int N)
{
    __shared__ float sIn[ROWS * RSTRIDE];
    __shared__ float sLo[72];   // zero-padded filter table: index j+32, j in [0,8) valid
    __shared__ float sHi[72];

    const int Nh   = N >> 1;
    const int tpb  = Nh / TILE_I;              // tiles per batch
    const int b    = blockIdx.x / tpb;
    const int i0   = (blockIdx.x % tpb) * TILE_I;
    const int tid  = threadIdx.x;

    // ---- build zero-padded filter tables (once per block) ----
    for (int t = tid; t < 72; t += 256) {
        int j  = t - 32;
        int jc = j < 0 ? 0 : (j > 7 ? 7 : j);
        float lo = kDecLo[jc], hi = kDecHi[jc];
        bool ok = (j >= 0) && (j < 8);
        sLo[t] = ok ? lo : 0.0f;
        sHi[t] = ok ? hi : 0.0f;
    }

    // ---- stage 72 input rows (circular) x 64 ch into LDS, float4 coalesced ----
    const float* inb = in + (size_t)b * N * C_CH;
    for (int idx = tid; idx < ROWS * (C_CH / 4); idx += 256) {
        int t  = idx >> 4;            // row 0..71
        int k4 = idx & 15;            // float4 within row
        int g  = 2 * i0 + t;
        if (g >= N) g -= N;           // 2*i0+71 < 2N always
        float4 v = reinterpret_cast<const float4*>(inb + (size_t)g * C_CH)[k4];
        *reinterpret_cast<float4*>(&sIn[t * RSTRIDE + (k4 << 2)]) = v;  // 16B aligned
    }
    __syncthreads();

    // ---- per-wave 16x16 tile pair (lo/hi) ----
    const int wave  = tid >> 5;
    const int lane  = tid & 31;
    const int isub  = wave >> 2;      // 0..1 : i-subtile
    const int csub  = wave & 3;       // 0..3 : c-subtile
    const int row16 = lane & 15;      // A: M row   | B: N col   | C: N col
    const int khalf = lane >> 4;      // 0 -> K{0,1}, 1 -> K{2,3}

    // A fragments: constant banded matrix M[i, col] = filt[col - 2i], col = 4q + 2*khalf + {0,1}
    v2f aLo[NCHUNK], aHi[NCHUNK];
#pragma unroll
    for (int q = 0; q < NCHUNK; ++q) {
        int col0 = 4 * q + 2 * khalf;
        int j0 = col0 - 2 * row16 + 32;          // in [2, 71]
        aLo[q][0] = sLo[j0];     aLo[q][1] = sLo[j0 + 1];
        aHi[q][0] = sHi[j0];     aHi[q][1] = sHi[j0 + 1];
    }

    // Accumulate: D = A(16x4) x B(4x16) + C, 10 chunks, shared B fragment for lo/hi
    v8f accLo = {};
    v8f accHi = {};
    const int rbase = 32 * isub;
    const int cbase = 16 * csub;
#pragma unroll
    for (int q = 0; q < NCHUNK; ++q) {
        int r = rbase + 4 * q + 2 * khalf;       // <= 32 + 36 + 2 = 70; +1 -> 71 (staged)
        v2f bfrag;
        bfrag[0] = sIn[(r)     * RSTRIDE + cbase + row16];
        bfrag[1] = sIn[(r + 1) * RSTRIDE + cbase + row16];
        accLo = __builtin_amdgcn_wmma_f32_16x16x4_f32(false, aLo[q], false, bfrag,
                                                      (short)0, accLo, false, false);
        accHi = __builtin_amdgcn_wmma_f32_16x16x4_f32(false, aHi[q], false, bfrag,
                                                      (short)0, accHi, false, false);
    }

    // ---- store: C/D layout: VGPR v -> M = v + 8*khalf, lane row16 -> channel ----
    const size_t obase = ((size_t)b * Nh + (i0 + 16 * isub)) * C_CH + cbase + row16;
#pragma unroll
    for (int v = 0; v < 8; ++v) {
        size_t off = obase + (size_t)(v + 8 * khalf) * C_CH;
        outA[off] = accLo[v];
        outD[off] = accHi[v];
    }
}

extern "C" void kernel_launch(void* const* d_in, const int* in_sizes, int n_in,
                              void* d_out, int out_size, void* d_ws, size_t ws_size,
                              hipStream_t stream) {
    (void)in_sizes; (void)n_in; (void)out_size; (void)ws_size;
    const float* x = (const float*)d_in[0];
    float* out = (float*)d_out;

    // workspace: A1 (32x4096x64 = 32MB), A2 (32x2048x64 = 16MB) - L2 resident
    float* A1 = (float*)d_ws;
    float* A2 = A1 + (size_t)32 * 4096 * 64;

    // output layout: A3 | D3 | D2 | D1
    const size_t SZ3 = (size_t)32 * 1024 * 64;   // 2,097,152
    float* A3 = out;
    float* D3 = out + SZ3;
    float* D2 = out + 2 * SZ3;
    float* D1 = out + 4 * SZ3;

    dim3 blk(256);
    // level 1: N=8192 -> 4096 tiles of 32 outputs across 32 batches
    dwt_level_wmma<<<dim3(32 * (4096 / TILE_I)), blk, 0, stream>>>(x,  A1, D1, 8192);
    // level 2: N=4096
    dwt_level_wmma<<<dim3(32 * (2048 / TILE_I)), blk, 0, stream>>>(A1, A2, D2, 4096);
    // level 3: N=2048
    dwt_level_wmma<<<dim3(32 * (1024 / TILE_I)), blk, 0, stream>>>(A2, A3, D3, 2048);
}